// GATWithEdgeFeatures_70428873720576
// MI455X (gfx1250) — compile-verified
//
#include <hip/hip_runtime.h>
#include <math.h>

#define N_PRO    60000
#define N_WAT    40000
#define NN       100000
#define NE       1000000
#define IN_DIM   21
#define HID      32
#define HEADS    4
#define HC       128          // HID*HEADS
#define EDGE_K   32
#define EDGE_DIM 32
#define FF_HID   128
#define RADIUS_MAX 5.0f

typedef __attribute__((ext_vector_type(2))) float    v2f;
typedef __attribute__((ext_vector_type(8))) float    v8f;
typedef __attribute__((ext_vector_type(4))) unsigned u32x4;
typedef __attribute__((ext_vector_type(8))) int      i32x8;
typedef __attribute__((ext_vector_type(4))) int      i32x4;

#if defined(__gfx1250__) && __has_builtin(__builtin_amdgcn_tensor_load_to_lds) && \
    __has_builtin(__builtin_amdgcn_s_wait_tensorcnt)
#define USE_TDM 1
#else
#define USE_TDM 0
#endif

// ---------- monotonic float<->u32 encoding for atomic float-max ----------
__device__ __forceinline__ unsigned fenc(float f) {
  unsigned u = __float_as_uint(f);
  return (u & 0x80000000u) ? ~u : (u | 0x80000000u);
}
__device__ __forceinline__ float fdec(unsigned e) {
  return (e & 0x80000000u) ? __uint_as_float(e ^ 0x80000000u)
                           : __uint_as_float(~e);
}

// ---------- generic zero fill ----------
__global__ void zero_u32(unsigned* __restrict__ p, int n) {
  int i = blockIdx.x * blockDim.x + threadIdx.x;
  if (i < n) p[i] = 0u;
}

// ---------- fold Wedge (32x128) with att_edge (4x32) -> [32][4], both convs ----------
__global__ void fold_edge_att(const float* __restrict__ Wedge1, const float* __restrict__ attE1,
                              const float* __restrict__ Wedge2, const float* __restrict__ attE2,
                              float* __restrict__ fold) {
  int t = threadIdx.x;                    // 256 threads, one block
  int conv = t >> 7, d = (t >> 2) & 31, h = t & 3;
  const float* Wg = conv ? Wedge2 : Wedge1;
  const float* at = conv ? attE2 : attE1;
  float s = 0.0f;
  #pragma unroll
  for (int c = 0; c < HID; ++c) s += Wg[d * HC + h * HID + c] * at[h * HID + c];
  fold[conv * EDGE_DIM * HEADS + d * HEADS + h] = s;
}

// ---------- per-edge: dist -> RBF -> edge_attr -> a_e1/a_e2 (edge_attr never stored) ----------
__global__ void edge_prep(const long long* __restrict__ ei,
                          const float* __restrict__ ppos, const float* __restrict__ wpos,
                          const float* __restrict__ gamma,
                          const float* __restrict__ We, const float* __restrict__ be,
                          const float* __restrict__ fold,
                          int* __restrict__ src32, int* __restrict__ dst32,
                          float* __restrict__ aE1, float* __restrict__ aE2) {
  int e = blockIdx.x * blockDim.x + threadIdx.x;
  if (e >= NE) return;
  int s = (int)ei[e], d = (int)ei[NE + e];
  src32[e] = s; dst32[e] = d;
  const float* ps = (s < N_PRO) ? (ppos + 3 * s) : (wpos + 3 * (s - N_PRO));
  const float* pd = (d < N_PRO) ? (ppos + 3 * d) : (wpos + 3 * (d - N_PRO));
  float dx = pd[0] - ps[0], dy = pd[1] - ps[1], dz = pd[2] - ps[2];
  float dist = sqrtf(dx * dx + dy * dy + dz * dz);
  float g = gamma[0];
  const float step = RADIUS_MAX / (float)(EDGE_K - 1);
  float rbf[EDGE_K];
  #pragma unroll
  for (int k = 0; k < EDGE_K; ++k) {
    float t = dist - step * (float)k;
    rbf[k] = expf(-g * t * t);
  }
  float a1[HEADS] = {0.f, 0.f, 0.f, 0.f};
  float a2[HEADS] = {0.f, 0.f, 0.f, 0.f};
  for (int c = 0; c < EDGE_DIM; ++c) {
    float ec = be[c];
    #pragma unroll
    for (int k = 0; k < EDGE_K; ++k) ec += rbf[k] * We[k * EDGE_DIM + c];
    #pragma unroll
    for (int h = 0; h < HEADS; ++h) {
      a1[h] += ec * fold[c * HEADS + h];
      a2[h] += ec * fold[EDGE_DIM * HEADS + c * HEADS + h];
    }
  }
  #pragma unroll
  for (int h = 0; h < HEADS; ++h) {
    aE1[e * HEADS + h] = a1[h];
    aE2[e * HEADS + h] = a2[h];
  }
}

// ---------- node feature embedding: feats @ Wf + bf (water rows = one-hot last slot) ----------
__global__ void node_embed(const float* __restrict__ pf, const float* __restrict__ Wf,
                           const float* __restrict__ bf, float* __restrict__ h0) {
  int t = blockIdx.x * blockDim.x + threadIdx.x;
  if (t >= NN * HID) return;
  int n = t >> 5, c = t & 31;
  float s = bf[c];
  if (n < N_PRO) {
    #pragma unroll
    for (int k = 0; k < IN_DIM; ++k) s += pf[n * IN_DIM + k] * Wf[k * HID + c];
  } else {
    s += Wf[(IN_DIM - 1) * HID + c];
  }
  h0[n * HID + c] = s;
}

// ---------- WMMA f32 GEMM: Out[nrows x 128] = Hin[nrows x DIN] @ W[DIN x 128] ----------
// block = 256 = 8 waves; each block does a 16-row stripe, each wave one 16-col tile.
// A tile (16 x DIN) is staged into LDS by the Tensor Data Mover (2D descriptor,
// tensor_dim1 = remaining rows so OOB rows zero-fill), falling back to a
// cooperative vector-load loop when the TDM builtin is unavailable.
// V_WMMA_F32_16X16X4_F32 layouts (ISA 7.12.2):
//   A 16x4:  lane&15 = M; (lane>=16 ? K=2,3 : K=0,1) in the v2f pair
//   B 4x16:  lane&15 = N; same K split
//   C/D:     VGPR r -> M = r (lanes 0-15) / r+8 (lanes 16-31), N = lane&15
template <int DIN>
__global__ __launch_bounds__(256) void gemm_wmma_f32(const float* __restrict__ Hin,
                                                     const float* __restrict__ W,
                                                     const float* __restrict__ bias, int relu,
                                                     float* __restrict__ Out, int nrows) {
  __shared__ alignas(16) float lds[16 * DIN];
  int tile = blockIdx.x;
  int tid = threadIdx.x;

#if USE_TDM
  if (tid < 32) {  // wave 0 issues one TDM 2D tile load: global[16 x DIN] -> LDS
    unsigned ldsOff = (unsigned)(uintptr_t)&lds[0];   // LDS aperture: addr[31:0] = offset
    unsigned long long ga =
        (unsigned long long)(uintptr_t)(Hin + (size_t)tile * 16 * DIN);
    unsigned rows = (unsigned)(nrows - tile * 16);    // >=1; rows past end zero-fill

    u32x4 g0;
    g0.x = 1u;                                          // count=1, user descriptor
    g0.y = ldsOff;                                      // lds_addr
    g0.z = (unsigned)(ga & 0xFFFFFFFFu);                // global_addr[31:0]
    g0.w = (unsigned)((ga >> 32) & 0x01FFFFFFu)         // global_addr[56:32]
         | (2u << 30);                                  // type=2 ("image")

    i32x8 g1;
    g1[0] = (int)(2u << 16);                            // wg_mask=0, data_size=4B
    g1[1] = (int)((DIN & 0xFFFFu) << 16);               // tensor_dim0 lo16
    g1[2] = (int)(((unsigned)DIN >> 16) | ((rows & 0xFFFFu) << 16)); // dim0 hi | dim1 lo
    g1[3] = (int)(((rows >> 16) & 0xFFFFu) | ((unsigned)DIN << 16)); // dim1 hi | tile_dim0
    g1[4] = 16;                                         // tile_dim1=16, tile_dim2=0
    g1[5] = DIN;                                        // tensor_dim0_stride lo32
    g1[6] = 0;                                          // stride hi16 | dim1_stride lo16
    g1[7] = 0;

    i32x4 z4 = {0, 0, 0, 0};
#if __clang_major__ >= 23
    i32x8 z8 = {0, 0, 0, 0, 0, 0, 0, 0};
    __builtin_amdgcn_tensor_load_to_lds(g0, g1, z4, z4, z8, 0);
#else
    __builtin_amdgcn_tensor_load_to_lds(g0, g1, z4, z4, 0);
#endif
    __builtin_amdgcn_s_wait_tensorcnt(0);
  }
  __syncthreads();
#else
  for (int i = tid; i < 16 * DIN; i += 256) {
    int r = i / DIN, c = i % DIN;
    int row = tile * 16 + r;
    lds[i] = (row < nrows) ? Hin[row * DIN + c] : 0.0f;
  }
  __syncthreads();
#endif

  int lane = tid & 31, wave = tid >> 5;
  int col  = wave * 16 + (lane & 15);
  int kb   = (lane >> 4) * 2;
  int rloc = lane & 15;

  v8f acc = {};
  #pragma unroll
  for (int k = 0; k < DIN; k += 4) {
    v2f a, b;
    a.x = lds[rloc * DIN + k + kb];
    a.y = lds[rloc * DIN + k + kb + 1];
    b.x = W[(k + kb) * HC + col];
    b.y = W[(k + kb + 1) * HC + col];
    acc = __builtin_amdgcn_wmma_f32_16x16x4_f32(false, a, false, b, (short)0, acc,
                                                false, false);
  }

  float bv = bias ? bias[col] : 0.0f;   // loop-invariant: hoisted out of the epilogue
  #pragma unroll
  for (int r = 0; r < 8; ++r) {
    int row = tile * 16 + ((lane < 16) ? r : r + 8);
    if (row < nrows) {
      float v = acc[r] + bv;
      if (relu) v = fmaxf(v, 0.0f);
      Out[row * HC + col] = v;
    }
  }
}

// ---------- per-node attention logits a_s, a_d from x ----------
__global__ void att_sd(const float* __restrict__ X, const float* __restrict__ att_s,
                       const float* __restrict__ att_d, float* __restrict__ aS,
                       float* __restrict__ aD) {
  int t = blockIdx.x * blockDim.x + threadIdx.x;
  if (t >= NN * HEADS) return;
  int n = t >> 2, h = t & 3;
  float s = 0.f, d = 0.f;
  #pragma unroll
  for (int c = 0; c < HID; ++c) {
    float v = X[n * HC + h * HID + c];
    s += v * att_s[h * HID + c];
    d += v * att_d[h * HID + c];
  }
  aS[t] = s;
  aD[t] = d;
}

// ---------- softmax pass 1: alpha = lrelu(...), atomic segment-max ----------
__global__ void edge_alpha_max(const int* __restrict__ src, const int* __restrict__ dst,
                               const float* __restrict__ aS, const float* __restrict__ aD,
                               const float* __restrict__ aE, float* __restrict__ exbuf,
                               unsigned* __restrict__ mEnc) {
  int t = blockIdx.x * blockDim.x + threadIdx.x;
  if (t >= NE * HEADS) return;
  int e = t >> 2, h = t & 3;
  float al = aS[src[e] * HEADS + h] + aD[dst[e] * HEADS + h] + aE[t];
  al = (al > 0.0f) ? al : 0.2f * al;          // leaky_relu(0.2)
  exbuf[t] = al;
  atomicMax(&mEnc[dst[e] * HEADS + h], fenc(al));
}

// ---------- softmax pass 2: ex = exp(alpha - m), atomic segment-sum ----------
__global__ void edge_exp_sum(const int* __restrict__ dst, float* __restrict__ exbuf,
                             const unsigned* __restrict__ mEnc, float* __restrict__ den) {
  int t = blockIdx.x * blockDim.x + threadIdx.x;
  if (t >= NE * HEADS) return;
  int e = t >> 2, h = t & 3;
  float m  = fdec(mEnc[dst[e] * HEADS + h]);
  float ex = expf(exbuf[t] - m);
  exbuf[t] = ex;
  atomicAdd(&den[dst[e] * HEADS + h], ex);
}

// ---------- pass 3: agg[dst] += x[src] * coef  (the bandwidth-bound core) ----------
__global__ void edge_aggregate(const int* __restrict__ src, const int* __restrict__ dst,
                               const float* __restrict__ exbuf, const float* __restrict__ den,
                               const float* __restrict__ X, float* __restrict__ Agg) {
  int t = blockIdx.x * blockDim.x + threadIdx.x;
  if (t >= NE * HEADS) return;
  int e = t >> 2, h = t & 3;
  int s = src[e], d = dst[e];
  float coef = exbuf[t] / (den[d * HEADS + h] + 1e-16f);
  const float4* __restrict__ xs4 =
      (const float4*)(X + (size_t)s * HC + h * HID);   // 16B-aligned (X is 256B-aligned)
  float* ag = Agg + (size_t)d * HC + h * HID;
  #pragma unroll
  for (int c = 0; c < HID / 4; ++c) {
    float4 v = xs4[c];                                  // global_load_b128 gather
    atomicAdd(&ag[4 * c + 0], v.x * coef);
    atomicAdd(&ag[4 * c + 1], v.y * coef);
    atomicAdd(&ag[4 * c + 2], v.z * coef);
    atomicAdd(&ag[4 * c + 3], v.w * coef);
  }
}

// ---------- elu(x + b) in place ----------
__global__ void elu_bias(float* __restrict__ A, const float* __restrict__ b) {
  int t = blockIdx.x * blockDim.x + threadIdx.x;
  if (t >= NN * HC) return;
  int c = t & (HC - 1);
  float v = A[t] + b[c];
  A[t] = (v > 0.0f) ? v : (expf(v) - 1.0f);
}

// ---------- final projection to 2 outputs, water rows only ----------
__global__ void mlp_out(const float* __restrict__ T, const float* __restrict__ Wm2,
                        const float* __restrict__ bm2, float* __restrict__ out) {
  int n = blockIdx.x * blockDim.x + threadIdx.x;
  if (n >= N_WAT) return;
  float a0 = bm2[0], a1 = bm2[1];
  #pragma unroll 4
  for (int k = 0; k < FF_HID; ++k) {
    float v = T[n * FF_HID + k];
    a0 += v * Wm2[k * 2 + 0];
    a1 += v * Wm2[k * 2 + 1];
  }
  out[n * 2 + 0] = a0;
  out[n * 2 + 1] = a1;
}

extern "C" void kernel_launch(void* const* d_in, const int* in_sizes, int n_in,
                              void* d_out, int out_size, void* d_ws, size_t ws_size,
                              hipStream_t stream) {
  const float*     ppos  = (const float*)d_in[0];
  const float*     wpos  = (const float*)d_in[1];
  const float*     pf    = (const float*)d_in[2];
  const long long* ei    = (const long long*)d_in[3];
  const float*     gamma = (const float*)d_in[4];
  const float* Wf  = (const float*)d_in[5];   const float* bf  = (const float*)d_in[6];
  const float* We  = (const float*)d_in[7];   const float* be  = (const float*)d_in[8];
  const float* W1  = (const float*)d_in[9];
  const float* as1 = (const float*)d_in[10];  const float* ad1 = (const float*)d_in[11];
  const float* Wg1 = (const float*)d_in[12];  const float* ae1w = (const float*)d_in[13];
  const float* b1  = (const float*)d_in[14];
  const float* W2  = (const float*)d_in[15];
  const float* as2 = (const float*)d_in[16];  const float* ad2 = (const float*)d_in[17];
  const float* Wg2 = (const float*)d_in[18];  const float* ae2w = (const float*)d_in[19];
  const float* b2  = (const float*)d_in[20];
  const float* Wm1 = (const float*)d_in[21];  const float* bm1 = (const float*)d_in[22];
  const float* Wm2 = (const float*)d_in[23];  const float* bm2 = (const float*)d_in[24];
  float* out = (float*)d_out;

  // ---- workspace layout ----
  char* ws = (char*)d_ws;
  size_t off = 0;
  auto alloc = [&](size_t bytes) -> void* {
    void* p = ws + off;
    off = (off + bytes + 255) & ~(size_t)255;
    return p;
  };
  int*      src32 = (int*)alloc(sizeof(int) * NE);
  int*      dst32 = (int*)alloc(sizeof(int) * NE);
  float*    aE1   = (float*)alloc(sizeof(float) * NE * HEADS);
  float*    aE2   = (float*)alloc(sizeof(float) * NE * HEADS);
  float*    h0    = (float*)alloc(sizeof(float) * NN * HID);
  float*    X     = (float*)alloc(sizeof(float) * NN * HC);   // x1 -> x2 -> mlp hidden
  float*    A     = (float*)alloc(sizeof(float) * NN * HC);   // agg1/h1 -> agg2/h2
  float*    aS    = (float*)alloc(sizeof(float) * NN * HEADS);
  float*    aD    = (float*)alloc(sizeof(float) * NN * HEADS);
  unsigned* mEnc  = (unsigned*)alloc(sizeof(unsigned) * NN * HEADS);
  float*    den   = (float*)alloc(sizeof(float) * NN * HEADS);
  float*    exb   = (float*)alloc(sizeof(float) * NE * HEADS);
  float*    fold  = (float*)alloc(sizeof(float) * 2 * EDGE_DIM * HEADS);

  const int B = 256;
  const int gEH   = (NE * HEADS + B - 1) / B;   // edge x head threads
  const int gNH   = (NN * HEADS + B - 1) / B;   // node x head threads
  const int gNC   = (NN * HC + B - 1) / B;      // node x channel threads
  const int gE    = (NE + B - 1) / B;
  const int gNE32 = (NN * HID + B - 1) / B;
  const int gTile = (NN + 15) / 16;

  // ---- preprocessing ----
  fold_edge_att<<<1, 256, 0, stream>>>(Wg1, ae1w, Wg2, ae2w, fold);
  edge_prep<<<gE, B, 0, stream>>>(ei, ppos, wpos, gamma, We, be, fold, src32, dst32, aE1, aE2);
  node_embed<<<gNE32, B, 0, stream>>>(pf, Wf, bf, h0);

  // ---- conv1 ----
  gemm_wmma_f32<HID><<<gTile, B, 0, stream>>>(h0, W1, nullptr, 0, X, NN);
  att_sd<<<gNH, B, 0, stream>>>(X, as1, ad1, aS, aD);
  zero_u32<<<gNH, B, 0, stream>>>(mEnc, NN * HEADS);
  zero_u32<<<gNH, B, 0, stream>>>((unsigned*)den, NN * HEADS);
  edge_alpha_max<<<gEH, B, 0, stream>>>(src32, dst32, aS, aD, aE1, exb, mEnc);
  edge_exp_sum<<<gEH, B, 0, stream>>>(dst32, exb, mEnc, den);
  zero_u32<<<gNC, B, 0, stream>>>((unsigned*)A, NN * HC);
  edge_aggregate<<<gEH, B, 0, stream>>>(src32, dst32, exb, den, X, A);
  elu_bias<<<gNC, B, 0, stream>>>(A, b1);                     // A = h1

  // ---- conv2 ----
  gemm_wmma_f32<HC><<<gTile, B, 0, stream>>>(A, W2, nullptr, 0, X, NN);   // X = x2
  att_sd<<<gNH, B, 0, stream>>>(X, as2, ad2, aS, aD);
  zero_u32<<<gNH, B, 0, stream>>>(mEnc, NN * HEADS);
  zero_u32<<<gNH, B, 0, stream>>>((unsigned*)den, NN * HEADS);
  edge_alpha_max<<<gEH, B, 0, stream>>>(src32, dst32, aS, aD, aE2, exb, mEnc);
  edge_exp_sum<<<gEH, B, 0, stream>>>(dst32, exb, mEnc, den);
  zero_u32<<<gNC, B, 0, stream>>>((unsigned*)A, NN * HC);
  edge_aggregate<<<gEH, B, 0, stream>>>(src32, dst32, exb, den, X, A);
  elu_bias<<<gNC, B, 0, stream>>>(A, b2);                     // A = h2

  // ---- MLP (water rows only) ----
  const int gTileW = (N_WAT + 15) / 16;
  gemm_wmma_f32<HC><<<gTileW, B, 0, stream>>>(A + (size_t)N_PRO * HC, Wm1, bm1, 1, X, N_WAT);
  mlp_out<<<(N_WAT + B - 1) / B, B, 0, stream>>>(X, Wm2, bm2, out);
}